// LiftSplatShoot_5772436046524
// MI455X (gfx1250) — compile-verified
//
#include <hip/hip_runtime.h>
#include <hip/hip_bf16.h>
#include <math.h>

#define DEPTH_BINS 39
#define FEATC 64
#define HFD 28
#define WFD 50
#define HBD 188
#define WBD 126
#define NCAM 6
#define NBATCH 2
#define NIMG (NBATCH*NCAM)
#define CHD (DEPTH_BINS + FEATC)        /* 103 */
#define MPOS (NBATCH*HBD*WBD)           /* 47376 == 16 * 2961 exactly */

typedef __attribute__((ext_vector_type(2))) float v2f;
typedef __attribute__((ext_vector_type(8))) float v8f;

// ---------------------------------------------------------------------------
// Per-camera transform: T = inv(car2cam) @ inv(K4).  car2cams is rigid, so
// inverse is [R^T | -R^T t].  invK3 = [[1/f,0,-cx/f],[0,1/f,-cy/f],[0,0,1]].
// ---------------------------------------------------------------------------
__global__ void compute_T_kernel(const float* __restrict__ K,
                                 const float* __restrict__ car2cams,
                                 float* __restrict__ T) {
  int i = threadIdx.x;
  if (i >= NIMG) return;
  const float* Ki = K + i * 9;
  const float* Mi = car2cams + i * 16;
  float f = Ki[0], cx = Ki[2], cy = Ki[5];
  float invf = 1.0f / f;
  float R[3][3], t[3];
  for (int r = 0; r < 3; ++r)
    for (int c = 0; c < 3; ++c) R[r][c] = Mi[c * 4 + r];   // R^T of car2cams
  float tc0 = Mi[3], tc1 = Mi[7], tc2 = Mi[11];
  for (int r = 0; r < 3; ++r)
    t[r] = -(R[r][0] * tc0 + R[r][1] * tc1 + R[r][2] * tc2);
  for (int r = 0; r < 3; ++r) {
    float a0 = R[r][0], a1 = R[r][1], a2 = R[r][2];
    T[i * 12 + r * 4 + 0] = a0 * invf;
    T[i * 12 + r * 4 + 1] = a1 * invf;
    T[i * 12 + r * 4 + 2] = -a0 * cx * invf - a1 * cy * invf + a2;
    T[i * 12 + r * 4 + 3] = t[r];
  }
}

// ---------------------------------------------------------------------------
// Fold BN into conv weights, k-pair interleaved for b64 B-fragment loads:
//   wr[((tap*(Cin/2) + k/2)*Cout + oc)*2 + (k&1)] = w[oc][k][tap] * s(oc)
//   br[oc] = (b-m)*s + beta,  s = g * rsqrt(v+eps).
// ---------------------------------------------------------------------------
__global__ void repack_kernel(const float* __restrict__ w, const float* __restrict__ b,
                              const float* __restrict__ g, const float* __restrict__ be,
                              const float* __restrict__ m, const float* __restrict__ v,
                              float* __restrict__ wr, float* __restrict__ br,
                              int Cin, int Cout) {
  int i = blockIdx.x * blockDim.x + threadIdx.x;
  int total = 9 * Cin * Cout;
  if (i < total) {
    int r    = i & 1;
    int pair = i >> 1;
    int oc   = pair % Cout;
    int rem  = pair / Cout;
    int kp   = rem % (Cin >> 1);
    int tap  = rem / (Cin >> 1);
    int ic   = 2 * kp + r;
    float s = g[oc] * rsqrtf(v[oc] + 1e-5f);
    wr[i] = w[((size_t)oc * Cin + ic) * 9 + tap] * s;
  }
  if (i < Cout) {
    float s = g[i] * rsqrtf(v[i] + 1e-5f);
    br[i] = (b[i] - m[i]) * s + be[i];
  }
}

__global__ void zero_kernel(float4* __restrict__ p, int n4) {
  int i = blockIdx.x * blockDim.x + threadIdx.x;
  if (i < n4) p[i] = make_float4(0.f, 0.f, 0.f, 0.f);
}

// ---------------------------------------------------------------------------
// Splat: one block (64 threads = 2 waves) per image pixel; thread t owns
// feature channel t.  Softmax over 39 depth logits via LDS; per depth bin,
// project, bin, coalesced fp32 atomics into channel-last BEV (L2 resident).
// ---------------------------------------------------------------------------
__global__ void splat_kernel(const float* __restrict__ head,
                             const float* __restrict__ T,
                             float* __restrict__ bev) {
  int pix = blockIdx.x;
  int x = pix % WFD;
  int r = pix / WFD;
  int h = r % HFD;
  int img = r / HFD;
  int b = img / NCAM;
  int t = threadIdx.x;

  __shared__ float logit[DEPTH_BINS];
  const float* hp = head + (size_t)img * CHD * HFD * WFD;
  if (t < DEPTH_BINS) logit[t] = hp[(t * HFD + h) * WFD + x];
  __syncthreads();

  float mx = -1e30f;
#pragma unroll
  for (int d = 0; d < DEPTH_BINS; ++d) mx = fmaxf(mx, logit[d]);
  float s = 0.f;
#pragma unroll
  for (int d = 0; d < DEPTH_BINS; ++d) s += __expf(logit[d] - mx);
  float inv_s = 1.0f / s;

  float feat = hp[((DEPTH_BINS + t) * HFD + h) * WFD + x];
  const float* Ti = T + img * 12;
  float T00 = Ti[0], T01 = Ti[1], T02 = Ti[2], T03 = Ti[3];
  float T10 = Ti[4], T11 = Ti[5], T12 = Ti[6], T13 = Ti[7];
  float u = x * 16.0f + 8.0f;
  float v = h * 16.0f + 8.0f;
  float* bb = bev + (size_t)b * HBD * WBD * FEATC + t;

  for (int d = 0; d < DEPTH_BINS; ++d) {
    float dep = 3.0f + 2.0f * (float)d;              // depth bin centers
    float wgt = __expf(logit[d] - mx) * inv_s;
    float px = u * dep, py = v * dep;
    float gx = T00 * px + T01 * py + T02 * dep + T03;
    float gy = T10 * px + T11 * py + T12 * dep + T13;
    int bx = (int)(gx / 0.8f);                        // trunc-to-zero == astype(int32)
    int by = (int)((gy + 50.4f) / 0.8f);
    if (bx >= 0 && bx < HBD && by >= 0 && by < WBD)
      atomicAdd(bb + ((size_t)bx * WBD + by) * FEATC, feat * wgt);
  }
}

// ---------------------------------------------------------------------------
// 3x3 conv as implicit-im2col GEMM on fp32 WMMA (16x16x4).
// Block = (COUT/16) waves; each wave owns a 16(M) x 16(N) output tile.
// MPOS is a multiple of 16 -> every tile is full, no bounds checks needed.
// A-panel (16 rows x CIN) staged in LDS per tap (padded stride, bank-safe).
// Boundary handling is branch-free: neighbor coords are clamped so the load
// address is always valid, then v_cndmask selects 0 for out-of-bounds taps.
// B fragments are single global b64 loads thanks to k-pair weight layout.
// ---------------------------------------------------------------------------
template <int CIN, int COUT>
__global__ void conv3x3_wmma_kernel(const float* __restrict__ in,
                                    const float* __restrict__ wr,
                                    const float* __restrict__ br,
                                    float* __restrict__ out) {
  constexpr int LDA = CIN + 2;                 // pad: rows land on distinct banks
  constexpr int NTHREADS = (COUT / 16) * 32;
  constexpr int IT = (16 * CIN) / NTHREADS;    // staged elements per thread
  __shared__ float tileA[16 * LDA];

  int tid = threadIdx.x;
  int wave = tid >> 5;
  int lane = tid & 31;
  int half = lane >> 4;
  int lcol = lane & 15;
  int m0 = blockIdx.x * 16;
  int tile_n = wave * 16;

  // Hoisted staging decode: fixed (row, chan) per thread across all 9 taps.
  int yv[IT], xv[IT], a0[IT], lo[IT];
#pragma unroll
  for (int it = 0; it < IT; ++it) {
    int i = tid + it * NTHREADS;
    int rr = i / CIN, c = i - rr * CIN;
    int pos = m0 + rr;                         // always < MPOS (16 | MPOS)
    int bI = pos / (HBD * WBD);
    int rem = pos - bI * (HBD * WBD);
    yv[it] = rem / WBD;
    xv[it] = rem - yv[it] * WBD;
    a0[it] = bI * (HBD * WBD * CIN) + c;
    lo[it] = rr * LDA + c;
  }

  v8f acc = {0.f, 0.f, 0.f, 0.f, 0.f, 0.f, 0.f, 0.f};

  for (int tap = 0; tap < 9; ++tap) {
    int dy = tap / 3 - 1, dx = tap % 3 - 1;
    __syncthreads();
#pragma unroll
    for (int it = 0; it < IT; ++it) {
      int yy = yv[it] + dy, xx = xv[it] + dx;
      bool inb = ((unsigned)yy < (unsigned)HBD) & ((unsigned)xx < (unsigned)WBD);
      int yyc = min(max(yy, 0), HBD - 1);      // clamp -> address always legal
      int xxc = min(max(xx, 0), WBD - 1);
      float val = in[a0[it] + yyc * (WBD * CIN) + xxc * CIN];  // unconditional
      tileA[lo[it]] = inb ? val : 0.f;         // branch-free select
    }
    __syncthreads();

    const v2f* wtap = (const v2f*)wr + (size_t)tap * (CIN / 2) * COUT;
    int koff = half * 2;
#pragma unroll 4
    for (int k0 = 0; k0 < CIN; k0 += 4) {
#if __has_builtin(__builtin_amdgcn_wmma_f32_16x16x4_f32)
      v2f a = *(const v2f*)&tileA[lcol * LDA + k0 + koff];
      v2f bf = wtap[(size_t)((k0 + koff) >> 1) * COUT + tile_n + lcol];
      acc = __builtin_amdgcn_wmma_f32_16x16x4_f32(
          /*neg_a=*/false, a, /*neg_b=*/false, bf,
          /*c_mod=*/(short)0, acc, /*reuse_a=*/false, /*reuse_b=*/false);
#else
      // Fallback with identical fragment semantics (no matrix pipe).
      for (int kk = 0; kk < 2; ++kk) {
        int k = k0 + koff + kk;
        v2f bp = wtap[(size_t)(k >> 1) * COUT + tile_n + lcol];
        float bv = (k & 1) ? bp.y : bp.x;
#pragma unroll
        for (int rr = 0; rr < 8; ++rr)
          acc[rr] += tileA[(rr + half * 8) * LDA + k] * bv;
      }
#endif
    }
  }

  float bias = br[tile_n + lcol];
  float* op = out + (size_t)(m0 + half * 8) * COUT + tile_n + lcol;
#pragma unroll
  for (int rr = 0; rr < 8; ++rr)
    op[(size_t)rr * COUT] = fmaxf(acc[rr] + bias, 0.0f);   // folded BN + ReLU
}

// ---------------------------------------------------------------------------
// Final 1x1 conv 64 -> 1 (no BN, no ReLU).
// ---------------------------------------------------------------------------
__global__ void conv1x1_kernel(const float* __restrict__ x,
                               const float* __restrict__ w4,
                               const float* __restrict__ b4,
                               float* __restrict__ out) {
  int pos = blockIdx.x * blockDim.x + threadIdx.x;
  if (pos >= MPOS) return;
  const float4* xp = (const float4*)(x + (size_t)pos * FEATC);
  const float4* wp = (const float4*)w4;
  float s = 0.f;
#pragma unroll
  for (int i = 0; i < FEATC / 4; ++i) {
    float4 a = xp[i], b = wp[i];
    s += a.x * b.x + a.y * b.y + a.z * b.z + a.w * b.w;
  }
  out[pos] = s + b4[0];
}

// ---------------------------------------------------------------------------
extern "C" void kernel_launch(void* const* d_in, const int* in_sizes, int n_in,
                              void* d_out, int out_size, void* d_ws, size_t ws_size,
                              hipStream_t stream) {
  (void)in_sizes; (void)n_in; (void)out_size; (void)ws_size;
  const float* head = (const float*)d_in[0];
  const float* K    = (const float*)d_in[1];
  const float* c2c  = (const float*)d_in[2];
  const float* w1 = (const float*)d_in[3];  const float* b1  = (const float*)d_in[4];
  const float* g1 = (const float*)d_in[5];  const float* be1 = (const float*)d_in[6];
  const float* m1 = (const float*)d_in[7];  const float* v1  = (const float*)d_in[8];
  const float* w2 = (const float*)d_in[9];  const float* b2  = (const float*)d_in[10];
  const float* g2 = (const float*)d_in[11]; const float* be2 = (const float*)d_in[12];
  const float* m2 = (const float*)d_in[13]; const float* v2  = (const float*)d_in[14];
  const float* w3 = (const float*)d_in[15]; const float* b3  = (const float*)d_in[16];
  const float* g3 = (const float*)d_in[17]; const float* be3 = (const float*)d_in[18];
  const float* m3 = (const float*)d_in[19]; const float* v3  = (const float*)d_in[20];
  const float* w4 = (const float*)d_in[21]; const float* b4  = (const float*)d_in[22];
  float* out = (float*)d_out;

  char* wsb = (char*)d_ws;
  size_t off = 0;
  auto alloc = [&](size_t bytes) -> char* {
    char* p = wsb + off;
    off = (off + bytes + 255) & ~(size_t)255;
    return p;
  };
  float* T   = (float*)alloc((size_t)NIMG * 12 * sizeof(float));
  float* bev = (float*)alloc((size_t)MPOS * FEATC * sizeof(float));
  float* x1  = (float*)alloc((size_t)MPOS * 128 * sizeof(float));
  float* x2  = (float*)alloc((size_t)MPOS * 128 * sizeof(float));
  float* x3  = (float*)alloc((size_t)MPOS * 64 * sizeof(float));
  float* w1r = (float*)alloc((size_t)9 * 64 * 128 * sizeof(float));
  float* b1r = (float*)alloc(128 * sizeof(float));
  float* w2r = (float*)alloc((size_t)9 * 128 * 128 * sizeof(float));
  float* b2r = (float*)alloc(128 * sizeof(float));
  float* w3r = (float*)alloc((size_t)9 * 128 * 64 * sizeof(float));
  float* b3r = (float*)alloc(64 * sizeof(float));

  compute_T_kernel<<<1, 32, 0, stream>>>(K, c2c, T);

  repack_kernel<<<(9 * 64 * 128 + 255) / 256, 256, 0, stream>>>(
      w1, b1, g1, be1, m1, v1, w1r, b1r, 64, 128);
  repack_kernel<<<(9 * 128 * 128 + 255) / 256, 256, 0, stream>>>(
      w2, b2, g2, be2, m2, v2, w2r, b2r, 128, 128);
  repack_kernel<<<(9 * 128 * 64 + 255) / 256, 256, 0, stream>>>(
      w3, b3, g3, be3, m3, v3, w3r, b3r, 128, 64);

  int n4 = MPOS * FEATC / 4;
  zero_kernel<<<(n4 + 255) / 256, 256, 0, stream>>>((float4*)bev, n4);

  splat_kernel<<<NIMG * HFD * WFD, 64, 0, stream>>>(head, T, bev);

  conv3x3_wmma_kernel<64, 128><<<MPOS / 16, 256, 0, stream>>>(bev, w1r, b1r, x1);
  conv3x3_wmma_kernel<128, 128><<<MPOS / 16, 256, 0, stream>>>(x1, w2r, b2r, x2);
  conv3x3_wmma_kernel<128, 64><<<MPOS / 16, 128, 0, stream>>>(x2, w3r, b3r, x3);

  conv1x1_kernel<<<(MPOS + 255) / 256, 256, 0, stream>>>(x3, w4, b4, out);
}